// BiAttn_25520695673286
// MI455X (gfx1250) — compile-verified
//
#include <hip/hip_runtime.h>

typedef float v2f __attribute__((ext_vector_type(2)));
typedef float v8f __attribute__((ext_vector_type(8)));

#define B_ 8
#define X_ 2048
#define Y_ 2048
#define H_ 1024

// ---------------------------------------------------------------------------
// Kernel 1: sq[b,y] = sum_h q[b,y,h] * Wq[h]   via V_WMMA_F32_16X16X4_F32
// Each wave owns 16 y-rows; K loops over H in 64-wide LDS tiles, 4 per WMMA.
// B operand = Wq broadcast across all 16 N columns, so every column of the
// 16x16 accumulator holds the same dot product (exact fp32).
// ---------------------------------------------------------------------------
__global__ __launch_bounds__(256) void biattn_sq_wmma(
    const float* __restrict__ q, const float* __restrict__ W,
    float* __restrict__ sq)
{
  __shared__ __align__(16) float wq[H_];
  __shared__ __align__(16) float qt[8][16 * 68];   // row stride 68 -> no bank conflicts

  const int tilesPerB = Y_ / 128;                  // 16
  const int b    = blockIdx.x / tilesPerB;
  const int ytil = blockIdx.x % tilesPerB;
  const int wave = threadIdx.x >> 5;
  const int lane = threadIdx.x & 31;

  for (int i = threadIdx.x; i < H_; i += 256) wq[i] = W[H_ + i];  // Wq = W[H:2H]
  __syncthreads();

  const int y0 = ytil * 128 + wave * 16;
  const float* qb = q + ((size_t)b * Y_ + y0) * H_;
  float* qtw = &qt[wave][0];

  const int m     = lane & 15;               // A-matrix row (M)
  const int khalf = (lane >> 4) << 1;        // lanes 0-15 -> K=0,1 ; lanes 16-31 -> K=2,3

  v8f acc = {};
  for (int k0 = 0; k0 < H_; k0 += 64) {
    // coalesced load of a 16x64 fp32 tile (1024 floats = 32 lanes x 8 float4)
    #pragma unroll
    for (int i = 0; i < 8; ++i) {
      int idx = i * 32 + lane;               // float4 slot 0..255
      int r   = idx >> 4;
      int c4  = idx & 15;
      float4 val = *reinterpret_cast<const float4*>(qb + (size_t)r * H_ + k0 + c4 * 4);
      *reinterpret_cast<float4*>(qtw + r * 68 + c4 * 4) = val;
    }
    __syncthreads();
    #pragma unroll
    for (int kk = 0; kk < 64; kk += 4) {
      v2f a, bb;
      a.x  = qtw[m * 68 + kk + khalf];
      a.y  = qtw[m * 68 + kk + khalf + 1];
      bb.x = wq[k0 + kk + khalf];            // broadcast across N
      bb.y = wq[k0 + kk + khalf + 1];
      acc = __builtin_amdgcn_wmma_f32_16x16x4_f32(
          /*neg_a=*/false, a, /*neg_b=*/false, bb,
          /*c_mod=*/(short)0, acc, /*reuse_a=*/false, /*reuse_b=*/false);
    }
    __syncthreads();
  }

  // D layout: VGPR r, lane L -> (M = r + 8*(L>>4), N = L&15); all N identical.
  if ((lane & 15) == 0) {
    const int mb = (lane >> 4) * 8;
    #pragma unroll
    for (int r = 0; r < 8; ++r)
      sq[(size_t)b * Y_ + y0 + mb + r] = acc[r];
  }
}

// ---------------------------------------------------------------------------
// Kernel 2: w[b,:] = softmax(sq[b,:]) ; also zero the o[b,:] accumulator.
// sk[b,x] and the scalar bias cancel under softmax (shift invariance).
// ---------------------------------------------------------------------------
__global__ __launch_bounds__(256) void biattn_softmax(
    const float* __restrict__ sq, float* __restrict__ w, float* __restrict__ o)
{
  __shared__ float red[256];
  const int b = blockIdx.x;
  const int t = threadIdx.x;
  const float* s = sq + (size_t)b * Y_;

  float lmax = -3.402823466e38f;
  for (int i = t; i < Y_; i += 256) lmax = fmaxf(lmax, s[i]);
  red[t] = lmax;
  __syncthreads();
  for (int off = 128; off > 0; off >>= 1) {
    if (t < off) red[t] = fmaxf(red[t], red[t + off]);
    __syncthreads();
  }
  const float mx = red[0];
  __syncthreads();

  float lsum = 0.f;
  for (int i = t; i < Y_; i += 256) lsum += __expf(s[i] - mx);
  red[t] = lsum;
  __syncthreads();
  for (int off = 128; off > 0; off >>= 1) {
    if (t < off) red[t] += red[t + off];
    __syncthreads();
  }
  const float inv = 1.0f / red[0];

  for (int i = t; i < Y_; i += 256) w[(size_t)b * Y_ + i] = __expf(s[i] - mx) * inv;
  for (int i = t; i < H_; i += 256) o[(size_t)b * H_ + i] = 0.f;
}

// ---------------------------------------------------------------------------
// Kernel 3: o[b,h] += sum_y w[b,y] * v[b,y,h]  (Y split across 32 blocks per b)
// Streams v (64 MB) fully coalesced; partial sums merged with f32 atomics.
// ---------------------------------------------------------------------------
__global__ __launch_bounds__(256) void biattn_gemv(
    const float* __restrict__ v, const float* __restrict__ w, float* o)
{
  const int YC = 64;
  const int b  = blockIdx.x / (Y_ / YC);
  const int yc = blockIdx.x % (Y_ / YC);
  const int t  = threadIdx.x;

  const float* vb = v + ((size_t)b * Y_ + (size_t)yc * YC) * H_;
  const float* wb = w + (size_t)b * Y_ + (size_t)yc * YC;

  float a0 = 0.f, a1 = 0.f, a2 = 0.f, a3 = 0.f;
  for (int y = 0; y < YC; ++y) {
    const float wy = wb[y];
    const float* row = vb + (size_t)y * H_;
    __builtin_prefetch(row + H_, 0, 1);      // global_prefetch_b8 (speculative, safe OOB)
    a0 = fmaf(wy, row[t],       a0);
    a1 = fmaf(wy, row[t + 256], a1);
    a2 = fmaf(wy, row[t + 512], a2);
    a3 = fmaf(wy, row[t + 768], a3);
  }
  float* ob = o + (size_t)b * H_;
  atomicAdd(ob + t,       a0);
  atomicAdd(ob + t + 256, a1);
  atomicAdd(ob + t + 512, a2);
  atomicAdd(ob + t + 768, a3);
}

// ---------------------------------------------------------------------------
// Kernel 4: out[b,x,h] = o[b,h]   (64 MB float4 broadcast store; o is L2-hot)
// ---------------------------------------------------------------------------
__global__ __launch_bounds__(256) void biattn_bcast(
    const float* __restrict__ o, float* __restrict__ out)
{
  const size_t idx4 = (size_t)blockIdx.x * 256 + threadIdx.x;  // float4 index
  const int    h4   = (int)(idx4 & (H_ / 4 - 1));              // 256 float4 per h-row
  const size_t bx   = idx4 >> 8;                               // b*X + x
  const int    b    = (int)(bx >> 11);                         // / X_
  const float4 val  = reinterpret_cast<const float4*>(o)[(size_t)b * (H_ / 4) + h4];
  reinterpret_cast<float4*>(out)[idx4] = val;
}

extern "C" void kernel_launch(void* const* d_in, const int* in_sizes, int n_in,
                              void* d_out, int out_size, void* d_ws, size_t ws_size,
                              hipStream_t stream)
{
  const float* q = (const float*)d_in[0];
  // d_in[1] = k : provably unused (sk[b,x] is constant along the softmax axis)
  const float* v = (const float*)d_in[2];
  const float* W = (const float*)d_in[3];
  // d_in[4] = scalar bias : also cancels under softmax

  float* ws = (float*)d_ws;
  float* sq = ws;                        // B*Y floats
  float* w  = ws + (size_t)B_ * Y_;      // B*Y floats
  float* o  = w  + (size_t)B_ * Y_;      // B*H floats   (total ~160 KB of ws)

  biattn_sq_wmma<<<B_ * (Y_ / 128), 256, 0, stream>>>(q, W, sq);
  biattn_softmax<<<B_, 256, 0, stream>>>(sq, w, o);
  biattn_gemv<<<B_ * (Y_ / 64), 256, 0, stream>>>(v, w, o);
  biattn_bcast<<<(int)(((size_t)B_ * X_ * H_ / 4) / 256), 256, 0, stream>>>(o, (float*)d_out);
}